// QuantumElectrochemicalSimulator_50122268344534
// MI455X (gfx1250) — compile-verified
//
#include <hip/hip_runtime.h>
#include <math.h>

typedef __attribute__((ext_vector_type(2))) float v2f;
typedef __attribute__((ext_vector_type(8))) float v8f;

#define S_DIM 4096
#define TILE  16

// ---------------------------------------------------------------------------
// Kernel 1: per-batch threshold = mean + 1.25*std of dist_sq over the full
// 4096x4096 plane. Fully separable -> O(S) reduction per batch, f64 accum.
// ---------------------------------------------------------------------------
__global__ __launch_bounds__(256) void qes_stats_kernel(
    const float* __restrict__ r, float* __restrict__ thr) {
  const int b = blockIdx.x;
  const float* rb = r + (size_t)b * S_DIM * 2;

  double la0 = 0, la1 = 0, lb0 = 0, lb1 = 0;
  double la00 = 0, la11 = 0, la01 = 0, lb00 = 0, lb11 = 0, lb01 = 0;
  for (int s = threadIdx.x; s < S_DIM; s += 256) {
    float r0 = rb[2 * s + 0], r1 = rb[2 * s + 1];
    float a0 = r0 * r0, a1 = r1 * r1;
    float u0 = 1.0f - r0, u1 = 1.0f - r1;
    float b0 = u0 * u0, b1 = u1 * u1;
    la0 += a0; la1 += a1; lb0 += b0; lb1 += b1;
    la00 += (double)a0 * a0; la11 += (double)a1 * a1; la01 += (double)a0 * a1;
    lb00 += (double)b0 * b0; lb11 += (double)b1 * b1; lb01 += (double)b0 * b1;
  }

  __shared__ double acc[10];
  if (threadIdx.x < 10) acc[threadIdx.x] = 0.0;
  __syncthreads();
  atomicAdd(&acc[0], la0);  atomicAdd(&acc[1], la1);
  atomicAdd(&acc[2], lb0);  atomicAdd(&acc[3], lb1);
  atomicAdd(&acc[4], la00); atomicAdd(&acc[5], la11); atomicAdd(&acc[6], la01);
  atomicAdd(&acc[7], lb00); atomicAdd(&acc[8], lb11); atomicAdd(&acc[9], lb01);
  __syncthreads();

  if (threadIdx.x == 0) {
    double S1 = acc[0] * acc[2] + acc[1] * acc[3];
    double S2 = acc[4] * acc[7] + acc[5] * acc[8] + 2.0 * acc[6] * acc[9];
    const double n = (double)S_DIM * (double)S_DIM;
    double mean = S1 / n;
    double var  = (S2 - S1 * S1 / n) / (n - 1.0);
    double sd   = sqrt(fmax(var, 0.0));
    sd = fmax(sd, 1e-6);
    thr[b] = (float)(mean + 1.25 * sd);
  }
}

// ---------------------------------------------------------------------------
// Kernel 2: one block per (batch, 16-row s-tile). r cached in LDS (32 KB).
// Per 16x16 tile: two V_WMMA_F32_16X16X4_F32 produce dist_sq and diff0 for
// 256 elements (rank-2 / rank-1 bilinear forms, shared B fragment).
// Phase 1 accumulates row sums (shuffle reduce + ds_add_f32).
// Phase 2 recomputes and writes normalized attention once (NT stores).
// ---------------------------------------------------------------------------
__global__ __launch_bounds__(256) void qes_attn_kernel(
    const float* __restrict__ r, const float* __restrict__ thr,
    float* __restrict__ out) {
  const int b    = blockIdx.y;
  const int s0   = blockIdx.x * TILE;
  const int tid  = threadIdx.x;
  const int lane = tid & 31;
  const int wave = tid >> 5;        // 0..7
  const int half = lane >> 4;       // 0: K0/K1 + rows 0-7, 1: K2/K3 + rows 8-15
  const int n    = lane & 15;

  __shared__ float2 rbuf[S_DIM];    // 32 KB: whole batch row of r
  __shared__ float  rowsum[TILE];
  __shared__ float  invRow[TILE];

  const float2* rb = (const float2*)(r + (size_t)b * S_DIM * 2);
  for (int i = tid; i < S_DIM; i += 256) rbuf[i] = rb[i];
  if (tid < TILE) rowsum[tid] = 0.0f;
  __syncthreads();

  const float threshold = thr[b];

  // A fragments, fixed for this block. A-matrix 16x4 f32: lane L supplies
  // row (L&15); lanes 0-15 -> K0,K1 ; lanes 16-31 -> K2,K3.
  //   dist : A = [rs0^2, rs1^2, 0, 0]
  //   diff0: A = [0, 0, rs0, 0]
  const float2 rs = rbuf[s0 + n];
  v2f aD, a0;
  if (half == 0) { aD.x = rs.x * rs.x; aD.y = rs.y * rs.y; a0.x = 0.f; a0.y = 0.f; }
  else           { aD.x = 0.f;         aD.y = 0.f;         a0.x = rs.x; a0.y = 0.f; }

  float racc[8];
#pragma unroll
  for (int v = 0; v < 8; ++v) racc[v] = 0.0f;

  // ---------------- phase 1: row sums ----------------
  for (int k = 0; k < 32; ++k) {
    const int t0 = (k * 8 + wave) * TILE;
    const float2 rt = rbuf[t0 + n];
    const float u0 = 1.0f - rt.x, u1 = 1.0f - rt.y;
    v2f bf;                       // shared B: K0=u0^2 K1=u1^2 | K2=u0 K3=u1
    if (half == 0) { bf.x = u0 * u0; bf.y = u1 * u1; }
    else           { bf.x = u0;      bf.y = u1;      }
    v8f z = {};
    v8f d  = __builtin_amdgcn_wmma_f32_16x16x4_f32(false, aD, false, bf,
                                                   (short)0, z, false, false);
    v8f f0 = __builtin_amdgcn_wmma_f32_16x16x4_f32(false, a0, false, bf,
                                                   (short)0, z, false, false);
#pragma unroll
    for (int v = 0; v < 8; ++v) {
      float dist = d[v];
      float c    = f0[v] * __frsqrt_rn(fmaxf(dist, 1e-30f)); // cos(atan2(y,x))
      float bias = 0.5f * (1.0f + c) * __expf(-dist);
      racc[v] += (dist <= threshold) ? bias : 0.0f;
    }
  }

  // reduce across the 16 lanes of each half (columns of the tile)
#pragma unroll
  for (int v = 0; v < 8; ++v) {
    float x = racc[v];
    x += __shfl_xor(x, 1);
    x += __shfl_xor(x, 2);
    x += __shfl_xor(x, 4);
    x += __shfl_xor(x, 8);
    racc[v] = x;
  }
  if (n == 0) {
#pragma unroll
    for (int v = 0; v < 8; ++v) atomicAdd(&rowsum[v + 8 * half], racc[v]);
  }
  __syncthreads();
  if (tid < TILE) invRow[tid] = 1.0f / (rowsum[tid] + 1e-8f);
  __syncthreads();

  float inv[8];
#pragma unroll
  for (int v = 0; v < 8; ++v) inv[v] = invRow[v + 8 * half];

  // ---------------- phase 2: recompute + normalized store ----------------
  float* outb = out + ((size_t)b * S_DIM + (size_t)s0) * S_DIM;
  for (int k = 0; k < 32; ++k) {
    const int t0 = (k * 8 + wave) * TILE;
    const float2 rt = rbuf[t0 + n];
    const float u0 = 1.0f - rt.x, u1 = 1.0f - rt.y;
    v2f bf;
    if (half == 0) { bf.x = u0 * u0; bf.y = u1 * u1; }
    else           { bf.x = u0;      bf.y = u1;      }
    v8f z = {};
    v8f d  = __builtin_amdgcn_wmma_f32_16x16x4_f32(false, aD, false, bf,
                                                   (short)0, z, false, false);
    v8f f0 = __builtin_amdgcn_wmma_f32_16x16x4_f32(false, a0, false, bf,
                                                   (short)0, z, false, false);
#pragma unroll
    for (int v = 0; v < 8; ++v) {
      float dist = d[v];
      float c    = f0[v] * __frsqrt_rn(fmaxf(dist, 1e-30f));
      float bias = 0.5f * (1.0f + c) * __expf(-dist);
      float att  = (dist <= threshold) ? bias * inv[v] : 0.0f;
      const int m = v + 8 * half;   // C/D layout: VGPR v -> row v (+8 upper half)
      __builtin_nontemporal_store(att, &outb[(size_t)m * S_DIM + t0 + n]);
    }
  }
}

extern "C" void kernel_launch(void* const* d_in, const int* in_sizes, int n_in,
                              void* d_out, int out_size, void* d_ws, size_t ws_size,
                              hipStream_t stream) {
  (void)in_sizes; (void)n_in; (void)out_size; (void)ws_size;
  const float* r   = (const float*)d_in[0];   // [2, 4096, 2] f32
  float*       out = (float*)d_out;           // [2, 4096, 4096] f32
  float*       thr = (float*)d_ws;            // 2 floats scratch

  qes_stats_kernel<<<2, 256, 0, stream>>>(r, thr);
  qes_attn_kernel<<<dim3(S_DIM / TILE, 2), 256, 0, stream>>>(r, thr, out);
}